// VGAEconv_12025908429199
// MI455X (gfx1250) — compile-verified
//
#include <hip/hip_runtime.h>
#include <hip/hip_bf16.h>

typedef __attribute__((ext_vector_type(2))) float v2f;
typedef __attribute__((ext_vector_type(8))) float v8f;

// ------------------------------------------------------------------
// 1) init: deg[i] = 1.0f (self loop), agg1 = 0
// ------------------------------------------------------------------
__global__ void k_init(float* __restrict__ deg, float* __restrict__ agg,
                       int n_nodes, int total) {
    int t = blockIdx.x * blockDim.x + threadIdx.x;
    if (t < total)   agg[t] = 0.0f;
    if (t < n_nodes) deg[t] = 1.0f;
}

// ------------------------------------------------------------------
// 2) in-degree accumulation over dst
// ------------------------------------------------------------------
__global__ void k_deg(const int* __restrict__ dst, float* __restrict__ deg, int E) {
    int e = blockIdx.x * blockDim.x + threadIdx.x;
    if (e < E) atomicAdd(&deg[dst[e]], 1.0f);
}

// ------------------------------------------------------------------
// 3) dinv = rsqrt(deg), in place
// ------------------------------------------------------------------
__global__ void k_dinv(float* __restrict__ deg, int n) {
    int i = blockIdx.x * blockDim.x + threadIdx.x;
    if (i < n) deg[i] = rsqrtf(deg[i]);
}

// ------------------------------------------------------------------
// 4) h0 = x @ W1   (M=100000, K=256, N=32) fp32 WMMA 16x16x4
//    one wave -> one 16-row tile, two N-tiles (N=0..15, 16..31), A reused
// ------------------------------------------------------------------
__global__ void k_gemm1(const float* __restrict__ x, const float* __restrict__ W1,
                        float* __restrict__ h0, int n_nodes) {
    const int lane = threadIdx.x & 31;
    const int half = lane >> 4;          // 0: K lanes {0,1}, 1: K lanes {2,3}
    const int l15  = lane & 15;
    const int wpb  = blockDim.x >> 5;
    const int n_tiles = (n_nodes + 15) >> 4;
    int tile = blockIdx.x * wpb + (threadIdx.x >> 5);
    const int tstride = gridDim.x * wpb;

    for (; tile < n_tiles; tile += tstride) {
        const int r0 = tile << 4;
        int mrow = r0 + l15;
        if (mrow >= n_nodes) mrow = n_nodes - 1;       // clamp (tile-level guard on store)
        const float* xrow = x + (size_t)mrow * 256 + half * 2;

        v8f acc0 = {0,0,0,0,0,0,0,0};
        v8f acc1 = {0,0,0,0,0,0,0,0};
        for (int k0 = 0; k0 < 256; k0 += 4) {
            v2f a;
            a.x = xrow[k0];                            // A[m][k0 + 2*half]
            a.y = xrow[k0 + 1];                        // A[m][k0 + 2*half + 1]
            const int krow = k0 + half * 2;
            v2f b0, b1;
            b0.x = W1[krow * 32 + l15];                // B[k][n],   n = 0..15
            b0.y = W1[(krow + 1) * 32 + l15];
            b1.x = W1[krow * 32 + 16 + l15];           // B[k][n+16]
            b1.y = W1[(krow + 1) * 32 + 16 + l15];
            acc0 = __builtin_amdgcn_wmma_f32_16x16x4_f32(false, a, false, b0,
                                                         (short)0, acc0, false, false);
            acc1 = __builtin_amdgcn_wmma_f32_16x16x4_f32(false, a, false, b1,
                                                         (short)0, acc1, false, false);
        }
        #pragma unroll
        for (int j = 0; j < 8; ++j) {                  // C/D: VGPR j -> rows j, j+8
            const int m = r0 + j + half * 8;
            if (m < n_nodes) {
                h0[(size_t)m * 32 + l15]      = acc0[j];
                h0[(size_t)m * 32 + 16 + l15] = acc1[j];
            }
        }
    }
}

// ------------------------------------------------------------------
// 5/7) edge scatter: one wave per edge, lane = feature channel (32)
//      agg[dst] += feat[src] * dinv[src]*dinv[dst]
// ------------------------------------------------------------------
__global__ void k_scatter(const int* __restrict__ src, const int* __restrict__ dst,
                          const float* __restrict__ dinv, const float* __restrict__ feat,
                          float* __restrict__ agg, int E) {
    const int lane = threadIdx.x & 31;
    const int e = (blockIdx.x * blockDim.x + threadIdx.x) >> 5;
    if (e >= E) return;
    const int s = src[e];
    const int d = dst[e];
    const float c = dinv[s] * dinv[d];
    const float v = feat[(size_t)s * 32 + lane] * c;
    atomicAdd(&agg[(size_t)d * 32 + lane], v);
}

// ------------------------------------------------------------------
// 6) h = relu(agg1 + h0*dinv^2 + b1) written over agg1 (bufB);
//    simultaneously zero bufA so it becomes the aggH accumulator
// ------------------------------------------------------------------
__global__ void k_relu(float* __restrict__ bufA /* h0 -> zeroed aggH */,
                       float* __restrict__ bufB /* agg1 -> h */,
                       const float* __restrict__ dinv, const float* __restrict__ b1,
                       int total) {
    int t = blockIdx.x * blockDim.x + threadIdx.x;
    if (t >= total) return;
    const int i = t >> 5;
    const int k = t & 31;
    const float di = dinv[i];
    float v = bufB[t] + bufA[t] * di * di + b1[k];
    bufB[t] = v > 0.0f ? v : 0.0f;
    bufA[t] = 0.0f;
}

// ------------------------------------------------------------------
// 8) A = aggH + h*dinv^2 (fused);  mu = A@Wm + bm; sigma = A@Wv + bv
//    (M=100000, K=32, N=16 twice) fp32 WMMA 16x16x4
// ------------------------------------------------------------------
__global__ void k_gemm2(const float* __restrict__ aggH, const float* __restrict__ h,
                        const float* __restrict__ dinv,
                        const float* __restrict__ Wm, const float* __restrict__ bm,
                        const float* __restrict__ Wv, const float* __restrict__ bv,
                        float* __restrict__ mu, float* __restrict__ sg, int n_nodes) {
    const int lane = threadIdx.x & 31;
    const int half = lane >> 4;
    const int l15  = lane & 15;
    const int wpb  = blockDim.x >> 5;
    const int n_tiles = (n_nodes + 15) >> 4;
    int tile = blockIdx.x * wpb + (threadIdx.x >> 5);
    const int tstride = gridDim.x * wpb;

    for (; tile < n_tiles; tile += tstride) {
        const int r0 = tile << 4;
        int mrow = r0 + l15;
        if (mrow >= n_nodes) mrow = n_nodes - 1;
        const float di = dinv[mrow];
        const float d2 = di * di;
        const size_t base = (size_t)mrow * 32 + half * 2;

        v8f acc0 = {0,0,0,0,0,0,0,0};
        v8f acc1 = {0,0,0,0,0,0,0,0};
        for (int k0 = 0; k0 < 32; k0 += 4) {
            const size_t idx = base + k0;
            v2f a;
            a.x = aggH[idx]     + h[idx]     * d2;     // fused self-loop add
            a.y = aggH[idx + 1] + h[idx + 1] * d2;
            const int krow = k0 + half * 2;
            v2f b0, b1;
            b0.x = Wm[krow * 16 + l15];
            b0.y = Wm[(krow + 1) * 16 + l15];
            b1.x = Wv[krow * 16 + l15];
            b1.y = Wv[(krow + 1) * 16 + l15];
            acc0 = __builtin_amdgcn_wmma_f32_16x16x4_f32(false, a, false, b0,
                                                         (short)0, acc0, false, false);
            acc1 = __builtin_amdgcn_wmma_f32_16x16x4_f32(false, a, false, b1,
                                                         (short)0, acc1, false, false);
        }
        #pragma unroll
        for (int j = 0; j < 8; ++j) {
            const int m = r0 + j + half * 8;
            if (m < n_nodes) {
                mu[(size_t)m * 16 + l15] = acc0[j] + bm[l15];
                sg[(size_t)m * 16 + l15] = acc1[j] + bv[l15];
            }
        }
    }
}

// ------------------------------------------------------------------
extern "C" void kernel_launch(void* const* d_in, const int* in_sizes, int n_in,
                              void* d_out, int out_size, void* d_ws, size_t ws_size,
                              hipStream_t stream) {
    const float* x  = (const float*)d_in[0];
    const int*   ei = (const int*)  d_in[1];
    const float* W1 = (const float*)d_in[2];
    const float* b1 = (const float*)d_in[3];
    const float* Wm = (const float*)d_in[4];
    const float* bm = (const float*)d_in[5];
    const float* Wv = (const float*)d_in[6];
    const float* bv = (const float*)d_in[7];

    const int n_nodes = in_sizes[0] / 256;
    const int E       = in_sizes[1] / 2;
    const int* src = ei;
    const int* dst = ei + E;

    float* ws   = (float*)d_ws;
    float* dinv = ws;                                  // n_nodes (deg -> dinv)
    float* bufA = ws + 102400;                         // n_nodes*32: h0, then aggH
    float* bufB = bufA + (size_t)n_nodes * 32;         // n_nodes*32: agg1, then h
    float* mu   = (float*)d_out;
    float* sg   = mu + (size_t)n_nodes * 16;

    const int total = n_nodes * 32;                    // 3.2M feature elements
    const int TB = 256;

    // 1) deg=1, agg1=0
    k_init<<<(total + TB - 1) / TB, TB, 0, stream>>>(dinv, bufB, n_nodes, total);
    // 2) in-degree atomics
    k_deg<<<(E + TB - 1) / TB, TB, 0, stream>>>(dst, dinv, E);
    // 3) dinv = rsqrt(deg)
    k_dinv<<<(n_nodes + TB - 1) / TB, TB, 0, stream>>>(dinv, n_nodes);
    // 4) h0 = x @ W1  (WMMA)
    {
        const int n_tiles = (n_nodes + 15) / 16;
        const int wpb = TB / 32;
        k_gemm1<<<(n_tiles + wpb - 1) / wpb, TB, 0, stream>>>(x, W1, bufA, n_nodes);
    }
    // 5) scatter pass 1: agg1 += h0[src]*coef
    {
        const long long threads = (long long)E * 32;
        k_scatter<<<(unsigned)((threads + TB - 1) / TB), TB, 0, stream>>>(
            src, dst, dinv, bufA, bufB, E);
    }
    // 6) h = relu(agg1 + h0*dinv^2 + b1); zero bufA for aggH
    k_relu<<<(total + TB - 1) / TB, TB, 0, stream>>>(bufA, bufB, dinv, b1, total);
    // 7) scatter pass 2: aggH += h[src]*coef  (single 32-dim pass shared by mu & sigma)
    {
        const long long threads = (long long)E * 32;
        k_scatter<<<(unsigned)((threads + TB - 1) / TB), TB, 0, stream>>>(
            src, dst, dinv, bufB, bufA, E);
    }
    // 8) mu/sigma = (aggH + h*dinv^2) @ [Wm|Wv] + bias  (WMMA)
    {
        const int n_tiles = (n_nodes + 15) / 16;
        const int wpb = TB / 32;
        k_gemm2<<<(n_tiles + wpb - 1) / wpb, TB, 0, stream>>>(
            bufA, bufB, dinv, Wm, bm, Wv, bv, mu, sg, n_nodes);
    }
}